// QOC_62594853372025
// MI455X (gfx1250) — compile-verified
//
#include <hip/hip_runtime.h>

// CDNA5 / gfx1250, wave32. Complex 8x8 time-ordered product via real 16x16
// embedding, chained V_WMMA_F32_16X16X4_F32, two independent sub-chains per
// wave for ILP (the problem is pure dependency-chain latency).

typedef float v2f __attribute__((ext_vector_type(2)));
typedef float v8f __attribute__((ext_vector_type(8)));

#define NSTEPS   8192
#define NCHUNK   64
#define CHUNKLEN (NSTEPS / NCHUNK)          // 128 steps per wave
#define HALFLEN  (CHUNKLEN / 2)             // 64 steps per sub-chain
#define HALFCMB  (NCHUNK / 2)               // 32 combines per sub-chain
#define DTSTEP   (1e-8f / 8192.0f)          // 1.220703125e-12

// ---------------------------------------------------------------------------
// D = A(16x4) x B(4x16) + C, f32, chained over K=16 in 4 slices.
// A layout: lane<16 -> row M=lane, K={4k,4k+1}; lane>=16 -> M=lane-16,
// K={4k+2,4k+3}.  B mirrors (rows in VGPRs, K=4k/4k+2 split by lane half).
// C/D: VGPR v = row v (lanes 0-15) / row v+8 (lanes 16-31), N = lane&15.
// ---------------------------------------------------------------------------
__device__ __forceinline__ v8f mat16_mul(const v2f a[4], const v2f b[4]) {
    v8f acc = {};
    acc = __builtin_amdgcn_wmma_f32_16x16x4_f32(false, a[0], false, b[0], (short)0, acc, false, false);
    acc = __builtin_amdgcn_wmma_f32_16x16x4_f32(false, a[1], false, b[1], (short)0, acc, false, false);
    acc = __builtin_amdgcn_wmma_f32_16x16x4_f32(false, a[2], false, b[2], (short)0, acc, false, false);
    acc = __builtin_amdgcn_wmma_f32_16x16x4_f32(false, a[3], false, b[3], (short)0, acc, false, false);
    return acc;
}

// Scatter C/D layout into a row-major 16x16 LDS tile.
__device__ __forceinline__ void c_to_lds(float* tile, int lane, const v8f& acc) {
    const int col   = lane & 15;
    const int rbase = (lane < 16) ? 0 : 8;
#pragma unroll
    for (int v = 0; v < 8; ++v)
        tile[(rbase + v) * 16 + col] = acc[v];
}

// Gather B fragments (row slices) from a row-major LDS tile.
__device__ __forceinline__ void lds_to_b(const float* tile, int lane, v2f b[4]) {
    const int col  = lane & 15;
    const int roff = (lane < 16) ? 0 : 2;
#pragma unroll
    for (int k = 0; k < 4; ++k) {
        b[k].x = tile[(4 * k + roff)     * 16 + col];
        b[k].y = tile[(4 * k + roff + 1) * 16 + col];
    }
}

// Gather A fragments (column slices) from a row-major LDS tile.
__device__ __forceinline__ void lds_to_a(const float* tile, int lane, v2f a[4]) {
    const int r  = lane & 15;
    const int co = ((lane >> 4) & 1) * 2;
#pragma unroll
    for (int k = 0; k < 4; ++k) {
        a[k].x = tile[r * 16 + 4 * k + co];
        a[k].y = tile[r * 16 + 4 * k + co + 1];
    }
}

// Identity matrix in B-fragment layout.
__device__ __forceinline__ void b_identity(int lane, v2f b[4]) {
    const int col  = lane & 15;
    const int roff = (lane < 16) ? 0 : 2;
#pragma unroll
    for (int k = 0; k < 4; ++k) {
        b[k].x = (col == 4 * k + roff)     ? 1.f : 0.f;
        b[k].y = (col == 4 * k + roff + 1) ? 1.f : 0.f;
    }
}

// Per-lane affine coefficients so that A-fragment element idx of
//   Uhat_i = I16 + dt*[[Im H, Re H], [-Re H, Im H]]   (U = I - i*dt*H; exact
//   at fp32 since ||H*dt|| ~ 3.5e-9, 2nd order term ~6e-18)
// is c0[idx] + a1[i]*c1[idx] + a2[i]*c2[idx].  H* are complex64 (interleaved).
__device__ __forceinline__ void build_coeffs(const float* Hd, const float* H1c,
                                             const float* H2c, int lane,
                                             float* c0, float* c1, float* c2) {
    const int r  = lane & 15;
    const int hi = (lane >> 4) & 1;
#pragma unroll
    for (int idx = 0; idx < 8; ++idx) {
        const int k = idx >> 1, j = idx & 1;
        const int c = 4 * k + 2 * hi + j;        // column 0..15
        const int rb = r & 7, cb = c & 7;
        const int qr = r >> 3, qc = c >> 3;
        float q_re, q_im;                         // embedding quadrant signs
        if (qr == qc)      { q_re = 0.f;  q_im = 1.f; }   // TL/BR: +dt*Im H
        else if (qr == 0)  { q_re = 1.f;  q_im = 0.f; }   // TR:   +dt*Re H
        else               { q_re = -1.f; q_im = 0.f; }   // BL:   -dt*Re H
        const int e = (rb * 8 + cb) * 2;          // complex64 interleaved
        c0[idx] = ((r == c) ? 1.f : 0.f) + DTSTEP * (q_re * Hd[e]  + q_im * Hd[e + 1]);
        c1[idx] =                          DTSTEP * (q_re * H1c[e] + q_im * H1c[e + 1]);
        c2[idx] =                          DTSTEP * (q_re * H2c[e] + q_im * H2c[e + 1]);
    }
}

// Build U_i A-fragments in registers from the affine coefficients.
__device__ __forceinline__ void build_a(const float* c0, const float* c1,
                                        const float* c2, float s1, float s2,
                                        v2f a[4]) {
#pragma unroll
    for (int k = 0; k < 4; ++k) {
        a[k].x = c0[2 * k]     + s1 * c1[2 * k]     + s2 * c2[2 * k];
        a[k].y = c0[2 * k + 1] + s1 * c1[2 * k + 1] + s2 * c2[2 * k + 1];
    }
}

// Merge two sub-chain results (both in C/D layout): R = hiP @ loP.
__device__ __forceinline__ v8f merge_chains(float* tile0, float* tile1, int lane,
                                            const v8f& loP, const v8f& hiP) {
    __syncthreads();
    c_to_lds(tile0, lane, loP);          // lo chain -> B operand
    c_to_lds(tile1, lane, hiP);          // hi chain -> A operand
    __syncthreads();
    v2f a[4], b[4];
    lds_to_b(tile0, lane, b);
    lds_to_a(tile1, lane, a);
    return mat16_mul(a, b);
}

// ---------------------------------------------------------------------------
// Kernel 1: each single-wave block folds CHUNKLEN steps as two interleaved
// independent 64-step sub-chains (2x ILP on the WMMA+LDS critical path),
// merges them, and writes one row-major 16x16 partial product.
// ---------------------------------------------------------------------------
__global__ void __launch_bounds__(32)
qoc_chunk_kernel(const float* __restrict__ a1, const float* __restrict__ a2,
                 const float* __restrict__ Hd, const float* __restrict__ H1c,
                 const float* __restrict__ H2c, float* __restrict__ partial) {
    __shared__ float tile0[256];
    __shared__ float tile1[256];
    const int lane  = threadIdx.x;
    const int chunk = blockIdx.x;

    float c0[8], c1[8], c2[8];
    build_coeffs(Hd, H1c, H2c, lane, c0, c1, c2);

    v2f bA[4], bB[4];
    b_identity(lane, bA);                // chain A carry = I (steps [0,64))
    b_identity(lane, bB);                // chain B carry = I (steps [64,128))

    v8f accA = {}, accB = {};
    const int baseA = chunk * CHUNKLEN;
    const int baseB = baseA + HALFLEN;
    for (int s = 0; s < HALFLEN; ++s) {
        const float s1a = a1[baseA + s], s2a = a2[baseA + s];  // uniform -> s_load
        const float s1b = a1[baseB + s], s2b = a2[baseB + s];
        v2f aA[4], aB[4];
        build_a(c0, c1, c2, s1a, s2a, aA);
        build_a(c0, c1, c2, s1b, s2b, aB);
        accA = mat16_mul(aA, bA);        // two independent WMMA chains
        accB = mat16_mul(aB, bB);
        if (s + 1 < HALFLEN) {           // shared D->B roundtrip, one wait
            __syncthreads();
            c_to_lds(tile0, lane, accA);
            c_to_lds(tile1, lane, accB);
            __syncthreads();
            lds_to_b(tile0, lane, bA);
            lds_to_b(tile1, lane, bB);
        }
    }

    const v8f acc = merge_chains(tile0, tile1, lane, accA, accB);

    float* dst = partial + chunk * 256;  // row-major 16x16 partial product
    const int col   = lane & 15;
    const int rbase = (lane < 16) ? 0 : 8;
#pragma unroll
    for (int v = 0; v < 8; ++v)
        dst[(rbase + v) * 16 + col] = acc[v];
}

// ---------------------------------------------------------------------------
// Kernel 2: single wave combines the NCHUNK partials (two interleaved
// sub-chains of 32), merges, and extracts the complex 8x8:
// Re = top-left block, Im = bottom-left block of the embedding.
// ---------------------------------------------------------------------------
__global__ void __launch_bounds__(32)
qoc_combine_kernel(const float* __restrict__ partial, float* __restrict__ out) {
    __shared__ float tile0[256];
    __shared__ float tile1[256];
    const int lane = threadIdx.x;
    const int r    = lane & 15;
    const int co   = ((lane >> 4) & 1) * 2;

    v2f bA[4], bB[4];
    b_identity(lane, bA);                // chain A: chunks [0,32)
    b_identity(lane, bB);                // chain B: chunks [32,64)

    v8f accA = {}, accB = {};
    for (int c = 0; c < HALFCMB; ++c) {
        const float* srcA = partial + c * 256;
        const float* srcB = partial + (HALFCMB + c) * 256;
        v2f aA[4], aB[4];
#pragma unroll
        for (int k = 0; k < 4; ++k) {    // partials as A-fragments (b64 loads)
            const int cc = 4 * k + co;
            aA[k].x = srcA[r * 16 + cc];
            aA[k].y = srcA[r * 16 + cc + 1];
            aB[k].x = srcB[r * 16 + cc];
            aB[k].y = srcB[r * 16 + cc + 1];
        }
        accA = mat16_mul(aA, bA);
        accB = mat16_mul(aB, bB);
        if (c + 1 < HALFCMB) {
            __syncthreads();
            c_to_lds(tile0, lane, accA);
            c_to_lds(tile1, lane, accB);
            __syncthreads();
            lds_to_b(tile0, lane, bA);
            lds_to_b(tile1, lane, bB);
        }
    }

    const v8f acc = merge_chains(tile0, tile1, lane, accA, accB);

    // complex64 output, interleaved re/im, row-major 8x8
#pragma unroll
    for (int v = 0; v < 8; ++v) {
        if (lane < 8)                      out[2 * (v * 8 + lane) + 0]        = acc[v]; // Re: M=v,   N=lane
        else if (lane >= 16 && lane < 24)  out[2 * (v * 8 + (lane - 16)) + 1] = acc[v]; // Im: M=v+8, N=lane-16
    }
}

// ---------------------------------------------------------------------------
extern "C" void kernel_launch(void* const* d_in, const int* in_sizes, int n_in,
                              void* d_out, int out_size, void* d_ws, size_t ws_size,
                              hipStream_t stream) {
    const float* a1  = (const float*)d_in[0];
    const float* a2  = (const float*)d_in[1];
    const float* Hd  = (const float*)d_in[2];   // complex64: 128 floats
    const float* H1c = (const float*)d_in[3];
    const float* H2c = (const float*)d_in[4];
    float* partial = (float*)d_ws;              // NCHUNK * 256 floats = 64 KB

    qoc_chunk_kernel<<<NCHUNK, 32, 0, stream>>>(a1, a2, Hd, H1c, H2c, partial);
    qoc_combine_kernel<<<1, 32, 0, stream>>>(partial, (float*)d_out);
}